// GRU_Decoder_pointer_generator_20194936226682
// MI455X (gfx1250) — compile-verified
//
#include <hip/hip_runtime.h>
#include <math.h>

#define V_    50000
#define E_    256
#define H_    512
#define S_    500
#define B_    64
#define OOV_  50
#define VEXT_ (V_ + OOV_)
#define XDIM_ 1280   /* E + 2H */
#define CDIM_ 1792   /* E + 3H */

typedef __attribute__((ext_vector_type(2))) float v2f;
typedef __attribute__((ext_vector_type(4))) float v4f;
typedef __attribute__((ext_vector_type(8))) float v8f;

// ---- workspace layout (float offsets) ----
#define WQ_OFF  0                          /* q       [B,H]   */
#define SC_OFF  (WQ_OFF + B_*H_)           /* alphas  [S,B]   */
#define CT_OFF  (SC_OFF + S_*B_)           /* context [B,2H]  */
#define EM_OFF  (CT_OFF + B_*2*H_)         /* embed   [B,E]   */
#define HN_OFF  (EM_OFF + B_*E_)           /* h_new   [B,H]   */
#define LN_OFF  (HN_OFF + B_*H_)           /* lin     [B,H]   */
#define PG_OFF  (LN_OFF + B_*H_)           /* p_gen   [B]     */
#define MX_OFF  (PG_OFF + B_)              /* row max [B]     */
#define SM_OFF  (MX_OFF + B_)              /* row sum [B]     */

__device__ __forceinline__ float sigmoidf_(float x) { return 1.0f / (1.0f + __expf(-x)); }

// ---------- q = prev_hidden @ Wq^T  [B,H]  (b-fastest: wave shares Wq row) ----------
__global__ void k_q(const float* __restrict__ prevh, const float* __restrict__ Wq,
                    float* __restrict__ q) {
  int idx = blockIdx.x * blockDim.x + threadIdx.x;   // o*B + b
  if (idx >= B_ * H_) return;
  int o = idx / B_, b = idx % B_;
  const float* hb = prevh + b * H_;
  const float* wr = Wq + (size_t)o * H_;
  float acc = 0.f;
  #pragma unroll 4
  for (int k = 0; k < H_; ++k) acc = fmaf(hb[k], wr[k], acc);
  q[b * H_ + o] = acc;
}

// ---------- embedding gather [B,E] ----------
__global__ void k_embed(const int* __restrict__ tok, const float* __restrict__ emb,
                        float* __restrict__ out) {
  int b = blockIdx.x, e = threadIdx.x;
  out[b * E_ + e] = emb[(size_t)tok[b] * E_ + e];
}

// ---------- scores[s,b] = We . tanh(q[b]+pk[s,b])  (wave per row, float4 lanes) ----------
__global__ void k_scores(const float* __restrict__ q, const float* __restrict__ pk,
                         const float* __restrict__ We, float* __restrict__ scores) {
  int wave = threadIdx.x >> 5;
  int lane = threadIdx.x & 31;
  int r = blockIdx.x * (blockDim.x >> 5) + wave;     // r = s*B + b
  if (r >= S_ * B_) return;
  int b = r % B_;
  const v4f* qb = (const v4f*)(q + b * H_);
  const v4f* pr = (const v4f*)(pk + (size_t)r * H_);
  const v4f* we = (const v4f*)We;
  float acc = 0.f;
  #pragma unroll
  for (int i = lane; i < H_ / 4; i += 32) {
    v4f p = __builtin_nontemporal_load(pr + i);   // proj_key streamed exactly once
    v4f qq = qb[i];
    v4f w  = we[i];
    acc += tanhf(qq.x + p.x) * w.x + tanhf(qq.y + p.y) * w.y +
           tanhf(qq.z + p.z) * w.z + tanhf(qq.w + p.w) * w.w;
  }
  #pragma unroll
  for (int off = 16; off > 0; off >>= 1) acc += __shfl_xor(acc, off, 32);
  if (lane == 0) scores[r] = acc;
}

// ---------- softmax over S (per batch column), in place ----------
__global__ void k_softmax_s(float* __restrict__ sc) {
  __shared__ float red[256];
  int b = blockIdx.x, tid = threadIdx.x;
  float m = -1e30f;
  for (int s = tid; s < S_; s += 256) m = fmaxf(m, sc[s * B_ + b]);
  red[tid] = m; __syncthreads();
  for (int st = 128; st > 0; st >>= 1) { if (tid < st) red[tid] = fmaxf(red[tid], red[tid + st]); __syncthreads(); }
  m = red[0]; __syncthreads();
  float sum = 0.f;
  for (int s = tid; s < S_; s += 256) sum += __expf(sc[s * B_ + b] - m);
  red[tid] = sum; __syncthreads();
  for (int st = 128; st > 0; st >>= 1) { if (tid < st) red[tid] += red[tid + st]; __syncthreads(); }
  float inv = 1.f / red[0];
  for (int s = tid; s < S_; s += 256) sc[s * B_ + b] = __expf(sc[s * B_ + b] - m) * inv;
}

// ---------- context[b,0:2H] = sum_s alpha[s,b]*enc[s,b,:]  (float4, NT stream) ----------
__global__ void k_context(const float* __restrict__ alphas, const float* __restrict__ ench,
                          float* __restrict__ ctx) {
  int d4 = threadIdx.x;          // 256 threads * float4 = 1024 = 2H
  int b = blockIdx.x;
  v4f acc = {0.f, 0.f, 0.f, 0.f};
  for (int s = 0; s < S_; ++s) {
    float a = alphas[s * B_ + b];
    v4f e = __builtin_nontemporal_load(
        (const v4f*)(ench + ((size_t)s * B_ + b) * (2 * H_)) + d4);
    acc += e * a;
  }
  *((v4f*)(ctx + b * 2 * H_) + d4) = acc;
}

// ---------- GRU cell (b-fastest: wave shares the three weight rows) ----------
__global__ void k_gru(const float* __restrict__ prevh, const float* __restrict__ embed,
                      const float* __restrict__ ctx, const float* __restrict__ Wih,
                      const float* __restrict__ Whh, const float* __restrict__ bih,
                      const float* __restrict__ bhh, float* __restrict__ hnew,
                      float* __restrict__ hout) {
  int idx = blockIdx.x * blockDim.x + threadIdx.x;   // j*B + b
  if (idx >= B_ * H_) return;
  int j = idx / B_, b = idx % B_;
  const float* wr = Wih + (size_t)j * XDIM_;
  const float* wz = Wih + (size_t)(H_ + j) * XDIM_;
  const float* wn = Wih + (size_t)(2 * H_ + j) * XDIM_;
  float ir = bih[j], iz = bih[H_ + j], in_ = bih[2 * H_ + j];
  const float* eb = embed + b * E_;
  #pragma unroll 4
  for (int k = 0; k < E_; ++k) {
    float x = eb[k];
    ir = fmaf(wr[k], x, ir); iz = fmaf(wz[k], x, iz); in_ = fmaf(wn[k], x, in_);
  }
  const float* cb = ctx + b * 2 * H_;
  #pragma unroll 4
  for (int k = 0; k < 2 * H_; ++k) {
    float x = cb[k];
    ir = fmaf(wr[E_ + k], x, ir); iz = fmaf(wz[E_ + k], x, iz); in_ = fmaf(wn[E_ + k], x, in_);
  }
  const float* ur = Whh + (size_t)j * H_;
  const float* uz = Whh + (size_t)(H_ + j) * H_;
  const float* un = Whh + (size_t)(2 * H_ + j) * H_;
  float hr = bhh[j], hz = bhh[H_ + j], hn = bhh[2 * H_ + j];
  const float* hb = prevh + b * H_;
  #pragma unroll 4
  for (int k = 0; k < H_; ++k) {
    float h = hb[k];
    hr = fmaf(ur[k], h, hr); hz = fmaf(uz[k], h, hz); hn = fmaf(un[k], h, hn);
  }
  float r = sigmoidf_(ir + hr);
  float z = sigmoidf_(iz + hz);
  float n = tanhf(in_ + r * hn);
  float hv = (1.f - z) * n + z * hb[j];
  hnew[b * H_ + j] = hv;
  hout[b * H_ + j] = hv;   // tuple output: h_new[None]
}

__device__ __forceinline__ float concat_feat(int b, int f, const float* __restrict__ embed,
                                             const float* __restrict__ hnew,
                                             const float* __restrict__ ctx) {
  if (f < E_) return embed[b * E_ + f];
  f -= E_;
  if (f < H_) return hnew[b * H_ + f];
  f -= H_;
  return ctx[b * 2 * H_ + f];
}

// ---------- lin[b,o] (b-fastest: wave shares lin_W row) ----------
__global__ void k_lin(const float* __restrict__ embed, const float* __restrict__ hnew,
                      const float* __restrict__ ctx, const float* __restrict__ linW,
                      const float* __restrict__ linb, float* __restrict__ lin) {
  int idx = blockIdx.x * blockDim.x + threadIdx.x;   // o*B + b
  if (idx >= B_ * H_) return;
  int o = idx / B_, b = idx % B_;
  const float* wr = linW + (size_t)o * CDIM_;
  float acc = linb[o];
  #pragma unroll 4
  for (int f = 0; f < CDIM_; ++f) acc = fmaf(wr[f], concat_feat(b, f, embed, hnew, ctx), acc);
  lin[b * H_ + o] = acc;
}

// ---------- p_gen[b] (block reduction) ----------
__global__ void k_pgen(const float* __restrict__ embed, const float* __restrict__ hnew,
                       const float* __restrict__ ctx, const float* __restrict__ pgW,
                       const float* __restrict__ pgb, float* __restrict__ pg) {
  __shared__ float red[256];
  int b = blockIdx.x, tid = threadIdx.x;
  float acc = 0.f;
  for (int f = tid; f < CDIM_; f += 256) acc = fmaf(pgW[f], concat_feat(b, f, embed, hnew, ctx), acc);
  red[tid] = acc; __syncthreads();
  for (int st = 128; st > 0; st >>= 1) { if (tid < st) red[tid] += red[tid + st]; __syncthreads(); }
  if (tid == 0) pg[b] = sigmoidf_(red[0] + pgb[0]);
}

// ---------- logits = lin @ out_W^T + out_b via V_WMMA_F32_16X16X4_F32 ----------
// One wave per 16-wide vocab tile computes ALL of M=64 (4 accumulators), so each
// out_W element is streamed from HBM exactly once (102 MB total, NT loads).
// A 16x4 f32 layout: lanes 0-15 hold {K=0,K=1}, lanes 16-31 hold {K=2,K=3}.
// C/D: lane<16 -> VGPR r = C[M=r][N=lane]; lane>=16 -> VGPR r = C[M=8+r][N=lane-16].
__global__ void k_logits_wmma(const float* __restrict__ lin, const float* __restrict__ outW,
                              const float* __restrict__ outb, float* __restrict__ out) {
  int lane = threadIdx.x;
  int half = lane >> 4;          // 0: K pair {0,1}, 1: K pair {2,3}
  int l    = lane & 15;
  int n0 = blockIdx.x * 16;      // vocab tile (3125 tiles, exact)
  const float* Brow = outW + (size_t)(n0 + l) * H_ + 2 * half;
  const float* A0   = lin  + (size_t)l * H_ + 2 * half;      // m-tiles 0..3
  const float* A1   = A0 + 16 * H_;
  const float* A2   = A0 + 32 * H_;
  const float* A3   = A0 + 48 * H_;
  v8f c0 = {}, c1 = {}, c2 = {}, c3 = {};
  for (int k = 0; k < H_; k += 4) {
    if ((k & 31) == 0) __builtin_prefetch(Brow + k + 256, 0, 1);  // one line ahead
    v2f bm = __builtin_nontemporal_load((const v2f*)(Brow + k));  // stream-once weights
    v2f a0 = *(const v2f*)(A0 + k);
    v2f a1 = *(const v2f*)(A1 + k);
    v2f a2 = *(const v2f*)(A2 + k);
    v2f a3 = *(const v2f*)(A3 + k);
    c0 = __builtin_amdgcn_wmma_f32_16x16x4_f32(false, a0, false, bm, (short)0, c0, false, false);
    c1 = __builtin_amdgcn_wmma_f32_16x16x4_f32(false, a1, false, bm, (short)0, c1, false, false);
    c2 = __builtin_amdgcn_wmma_f32_16x16x4_f32(false, a2, false, bm, (short)0, c2, false, false);
    c3 = __builtin_amdgcn_wmma_f32_16x16x4_f32(false, a3, false, bm, (short)0, c3, false, false);
  }
  float bias = outb[n0 + l];
  int col = n0 + l;
  #pragma unroll
  for (int r = 0; r < 8; ++r) {
    int m = half * 8 + r;
    out[(size_t)(m)      * VEXT_ + col] = c0[r] + bias;
    out[(size_t)(m + 16) * VEXT_ + col] = c1[r] + bias;
    out[(size_t)(m + 32) * VEXT_ + col] = c2[r] + bias;
    out[(size_t)(m + 48) * VEXT_ + col] = c3[r] + bias;
  }
}

// ---------- vocab softmax stats per batch row (single-pass online) ----------
__global__ void k_vstats(const float* __restrict__ logits, float* __restrict__ mx,
                         float* __restrict__ sm) {
  __shared__ float rm[256], rs[256];
  int b = blockIdx.x, tid = threadIdx.x;
  const float* row = logits + (size_t)b * VEXT_;
  float m = -1e30f, s = 0.f;
  for (int v = tid; v < V_; v += 256) {
    float x = row[v];
    float nm = fmaxf(m, x);
    s = s * __expf(m - nm) + __expf(x - nm);
    m = nm;
  }
  rm[tid] = m; rs[tid] = s; __syncthreads();
  for (int st = 128; st > 0; st >>= 1) {
    if (tid < st) {
      float m2 = rm[tid + st], s2 = rs[tid + st];
      float nm = fmaxf(rm[tid], m2);
      rs[tid] = rs[tid] * __expf(rm[tid] - nm) + s2 * __expf(m2 - nm);
      rm[tid] = nm;
    }
    __syncthreads();
  }
  if (tid == 0) { mx[b] = rm[0]; sm[b] = rs[0]; }
}

// ---------- ext[b,v] = p_gen*softmax + 1e-12 (in place; v>=V gets 1e-12) ----------
__global__ void k_extfill(float* __restrict__ out, const float* __restrict__ pg,
                          const float* __restrict__ mx, const float* __restrict__ sm) {
  int idx = blockIdx.x * blockDim.x + threadIdx.x;
  if (idx >= B_ * VEXT_) return;
  int b = idx / VEXT_, v = idx % VEXT_;
  float val = 1e-12f;
  if (v < V_) val += pg[b] * __expf(out[idx] - mx[b]) / sm[b];
  out[idx] = val;
}

// ---------- deterministic parallel scatter-add (owner-combine, no atomics) ----------
// One block per batch row. Stage (vocab, val) pairs in LDS; the first occurrence
// of each vocab id sums its duplicate group in ascending-s order (matching the
// reference's serial scatter semantics) and does one conflict-free global RMW.
__global__ void k_scatter(float* __restrict__ out, const float* __restrict__ alphas,
                          const float* __restrict__ pg, const int* __restrict__ extv) {
  __shared__ int   vid[S_];
  __shared__ float val[S_];
  int b = blockIdx.x, t = threadIdx.x;     // blockDim = 512
  if (t < S_) {
    vid[t] = extv[b * S_ + t];
    val[t] = (1.f - pg[b]) * alphas[t * B_ + b];
  }
  __syncthreads();
  if (t < S_) {
    int v = vid[t];
    bool owner = true;
    for (int j = 0; j < t; ++j)
      if (vid[j] == v) { owner = false; break; }
    if (owner) {
      float acc = val[t];
      for (int j = t + 1; j < S_; ++j)
        if (vid[j] == v) acc += val[j];
      out[(size_t)b * VEXT_ + v] += acc;
    }
  }
}

// ---------- final log, in place ----------
__global__ void k_log(float* __restrict__ out) {
  int idx = blockIdx.x * blockDim.x + threadIdx.x;
  if (idx >= B_ * VEXT_) return;
  out[idx] = logf(out[idx]);
}

extern "C" void kernel_launch(void* const* d_in, const int* in_sizes, int n_in,
                              void* d_out, int out_size, void* d_ws, size_t ws_size,
                              hipStream_t stream) {
  (void)in_sizes; (void)n_in; (void)out_size; (void)ws_size;
  const int*   tok   = (const int*)  d_in[0];
  const float* ench  = (const float*)d_in[1];
  const float* pk    = (const float*)d_in[2];
  /* d_in[3] src_mask: unused (as in reference) */
  const float* prevh = (const float*)d_in[4];
  const int*   extv  = (const int*)  d_in[5];
  /* d_in[6] batch_max_oov: compile-time OOV_ */
  const float* emb   = (const float*)d_in[7];
  const float* Wq    = (const float*)d_in[8];
  const float* We    = (const float*)d_in[9];
  const float* Wih   = (const float*)d_in[10];
  const float* Whh   = (const float*)d_in[11];
  const float* bih   = (const float*)d_in[12];
  const float* bhh   = (const float*)d_in[13];
  const float* linW  = (const float*)d_in[14];
  const float* linb  = (const float*)d_in[15];
  const float* pgW   = (const float*)d_in[16];
  const float* pgb   = (const float*)d_in[17];
  const float* outW  = (const float*)d_in[18];
  const float* outb  = (const float*)d_in[19];
  float* out = (float*)d_out;
  float* ws  = (float*)d_ws;

  k_q        <<<(B_*H_+255)/256, 256, 0, stream>>>(prevh, Wq, ws + WQ_OFF);
  k_embed    <<<B_, E_, 0, stream>>>(tok, emb, ws + EM_OFF);
  k_scores   <<<(S_*B_)/8, 256, 0, stream>>>(ws + WQ_OFF, pk, We, ws + SC_OFF);
  k_softmax_s<<<B_, 256, 0, stream>>>(ws + SC_OFF);
  k_context  <<<B_, 256, 0, stream>>>(ws + SC_OFF, ench, ws + CT_OFF);
  k_gru      <<<(B_*H_+255)/256, 256, 0, stream>>>(prevh, ws + EM_OFF, ws + CT_OFF,
                                                   Wih, Whh, bih, bhh,
                                                   ws + HN_OFF, out + (size_t)B_ * VEXT_);
  k_lin      <<<(B_*H_+255)/256, 256, 0, stream>>>(ws + EM_OFF, ws + HN_OFF, ws + CT_OFF,
                                                   linW, linb, ws + LN_OFF);
  k_pgen     <<<B_, 256, 0, stream>>>(ws + EM_OFF, ws + HN_OFF, ws + CT_OFF, pgW, pgb, ws + PG_OFF);
  k_logits_wmma<<<V_/16, 32, 0, stream>>>(ws + LN_OFF, outW, outb, out);
  k_vstats   <<<B_, 256, 0, stream>>>(out, ws + MX_OFF, ws + SM_OFF);
  k_extfill  <<<(B_*VEXT_+255)/256, 256, 0, stream>>>(out, ws + PG_OFF, ws + MX_OFF, ws + SM_OFF);
  k_scatter  <<<B_, 512, 0, stream>>>(out, ws + SC_OFF, ws + PG_OFF, extv);
  k_log      <<<(B_*VEXT_+255)/256, 256, 0, stream>>>(out);
}